// BoxListNMS_49658411876611
// MI455X (gfx1250) — compile-verified
//
#include <hip/hip_runtime.h>
#include <stdint.h>

#define NBOX  8192
#define WORDS (NBOX / 32)        // 256 u32 words per suppression-mask row
#define IOU_T 0.5f
#define MAXP  1000

// ---------------- async global->LDS (gfx1250) helpers, guarded ----------------
// Probe result (round 1): builtin exists; _b32 flavor takes int* params.
#if defined(__HIP_DEVICE_COMPILE__) && __has_builtin(__builtin_amdgcn_global_load_async_to_lds_b32)
#define USE_ASYNC_LDS 1
typedef __attribute__((address_space(1))) int* gi_ptr_t;   // global src
typedef __attribute__((address_space(3))) int* li_ptr_t;   // LDS dst
__device__ static inline gi_ptr_t as_global_i(const void* p) {
  return (gi_ptr_t)(unsigned long long)p;
}
__device__ static inline li_ptr_t as_lds_i(void* p) {
  // generic LDS address: low 32 bits are the LDS offset on gfx1250
  return (li_ptr_t)(unsigned int)(unsigned long long)p;
}
#else
#define USE_ASYNC_LDS 0
#endif

// ---------------- workspace layout (on d_ws) ----------------
// float sx1[N], sy1[N], sx2[N], sy2[N], ss[N], sar[N]   (6*N floats = 192KB)
// u32   mask[N * WORDS]                                  (8MB)
// u32   keep[N]                                          (32KB)

// =============== kernel 1: in-LDS bitonic argsort (descending by score) =======
__global__ __launch_bounds__(1024)
void nms_sort(const float* __restrict__ boxes, const float* __restrict__ scores,
              float* __restrict__ ws) {
  __shared__ float          skey[NBOX];   // 32KB
  __shared__ unsigned short sidx[NBOX];   // 16KB
  const int tid = threadIdx.x;

  for (int i = tid; i < NBOX; i += 1024) {
    skey[i] = scores[i];
    sidx[i] = (unsigned short)i;
  }

  for (unsigned k = 2; k <= NBOX; k <<= 1) {
    for (unsigned j = k >> 1; j > 0; j >>= 1) {
      __syncthreads();
      for (unsigned i = tid; i < NBOX; i += 1024) {
        unsigned ixj = i ^ j;
        if (ixj > i) {
          float a = skey[i], b = skey[ixj];
          bool up = ((i & k) == 0);
          bool sw = up ? (a < b) : (a > b);   // descending overall
          if (sw) {
            skey[i] = b; skey[ixj] = a;
            unsigned short t = sidx[i]; sidx[i] = sidx[ixj]; sidx[ixj] = t;
          }
        }
      }
    }
  }
  __syncthreads();

  float* sx1 = ws;
  float* sy1 = ws + NBOX;
  float* sx2 = ws + 2 * NBOX;
  float* sy2 = ws + 3 * NBOX;
  float* ss  = ws + 4 * NBOX;
  float* sar = ws + 5 * NBOX;
  for (int p = tid; p < NBOX; p += 1024) {
    int id = sidx[p];
    float x1 = boxes[4 * id + 0];
    float y1 = boxes[4 * id + 1];
    float x2 = boxes[4 * id + 2];
    float y2 = boxes[4 * id + 3];
    sx1[p] = x1; sy1[p] = y1; sx2[p] = x2; sy2[p] = y2;
    ss[p]  = skey[p];
    sar[p] = (x2 - x1) * (y2 - y1);
  }
}

// =============== kernel 2: pairwise suppression bit-mask ======================
// grid: (NBOX/128, WORDS), block: 128. Each block handles 128 rows x 32 cols.
__global__ __launch_bounds__(128)
void nms_mask(const float* __restrict__ ws, uint32_t* __restrict__ mask) {
  const int c0    = blockIdx.y * 32;
  const int rbase = blockIdx.x * 128;
  const int r     = rbase + threadIdx.x;

  // triangular early-out: suppression needs col > row; if every col <= every
  // row in this tile, the whole tile is zero.
  if (c0 + 31 <= rbase) {
    mask[(size_t)r * WORDS + blockIdx.y] = 0u;
    return;
  }

  const float* sx1 = ws;
  const float* sy1 = ws + NBOX;
  const float* sx2 = ws + 2 * NBOX;
  const float* sy2 = ws + 3 * NBOX;
  const float* sar = ws + 5 * NBOX;

  __shared__ float cx1[32], cy1[32], cx2[32], cy2[32], car[32];
  const int t = threadIdx.x;

#if USE_ASYNC_LDS
  if (t < 32) {
    const int c = c0 + t;
    __builtin_amdgcn_global_load_async_to_lds_b32(as_global_i(sx1 + c), as_lds_i(&cx1[t]), 0, 0);
    __builtin_amdgcn_global_load_async_to_lds_b32(as_global_i(sy1 + c), as_lds_i(&cy1[t]), 0, 0);
    __builtin_amdgcn_global_load_async_to_lds_b32(as_global_i(sx2 + c), as_lds_i(&cx2[t]), 0, 0);
    __builtin_amdgcn_global_load_async_to_lds_b32(as_global_i(sy2 + c), as_lds_i(&cy2[t]), 0, 0);
    __builtin_amdgcn_global_load_async_to_lds_b32(as_global_i(sar + c), as_lds_i(&car[t]), 0, 0);
  }
  asm volatile("s_wait_asynccnt 0" ::: "memory");
  __syncthreads();
#else
  if (t < 32) {
    const int c = c0 + t;
    cx1[t] = sx1[c]; cy1[t] = sy1[c];
    cx2[t] = sx2[c]; cy2[t] = sy2[c];
    car[t] = sar[c];
  }
  __syncthreads();
#endif

  const float rx1 = sx1[r], ry1 = sy1[r], rx2 = sx2[r], ry2 = sy2[r], ra = sar[r];

  uint32_t bits = 0;
#pragma unroll
  for (int c = 0; c < 32; ++c) {
    float xx1 = fmaxf(rx1, cx1[c]);
    float yy1 = fmaxf(ry1, cy1[c]);
    float xx2 = fminf(rx2, cx2[c]);
    float yy2 = fminf(ry2, cy2[c]);
    float iw  = fmaxf(xx2 - xx1, 0.0f);
    float ih  = fmaxf(yy2 - yy1, 0.0f);
    float inter = iw * ih;
    float uni   = ra + car[c] - inter;
    // inter/uni > 0.5  <=>  inter > 0.5*uni  (uni > 0 always: wh >= 1)
    bool sup = (inter > IOU_T * uni) && ((c0 + c) > r);
    bits |= ((uint32_t)sup) << c;
  }
  mask[(size_t)r * WORDS + blockIdx.y] = bits;
}

// =============== kernel 3: serial greedy reduce (single block) ================
__global__ __launch_bounds__(256)
void nms_reduce(const uint32_t* __restrict__ mask, uint32_t* __restrict__ keep) {
  __shared__ uint32_t removed_lds[WORDS];   // 1KB suppression state
  const int t = threadIdx.x;                // 0..255, owns word t
  removed_lds[t] = 0u;
  __syncthreads();

  int cnt = 0;  // uniform across threads
  for (int w = 0; w < WORDS; ++w) {
    uint32_t cur = removed_lds[w];          // uniform broadcast read
    for (int b = 0; b < 32; ++b) {
      const int i = w * 32 + b;
      if (i + 1 < NBOX) {
        // pull the next row's mask words toward L2/L0 while we decide
        __builtin_prefetch(&mask[(size_t)(i + 1) * WORDS + t], 0, 1);
      }
      if (((cur >> b) & 1u) == 0u) {
        // row i survives: it suppresses (cap is applied post-hoc, like ref)
        ++cnt;
        if (t == 0) keep[i] = (cnt <= MAXP) ? 1u : 0u;
        removed_lds[t] |= mask[(size_t)i * WORDS + t];
        __syncthreads();
        cur = removed_lds[w];
      } else {
        if (t == 0) keep[i] = 0u;
      }
    }
  }
}

// =============== kernel 4: write output =======================================
__global__ __launch_bounds__(256)
void nms_out(const float* __restrict__ ws, const uint32_t* __restrict__ keep,
             float* __restrict__ out) {
  const int i = blockIdx.x * 256 + threadIdx.x;
  if (i >= NBOX) return;
  const float k = keep[i] ? 1.0f : 0.0f;
  out[i * 5 + 0] = ws[i] * k;               // x1
  out[i * 5 + 1] = ws[NBOX + i] * k;        // y1
  out[i * 5 + 2] = ws[2 * NBOX + i] * k;    // x2
  out[i * 5 + 3] = ws[3 * NBOX + i] * k;    // y2
  out[i * 5 + 4] = ws[4 * NBOX + i] * k;    // score
}

// =============== launcher =====================================================
extern "C" void kernel_launch(void* const* d_in, const int* in_sizes, int n_in,
                              void* d_out, int out_size, void* d_ws, size_t ws_size,
                              hipStream_t stream) {
  const float* boxes  = (const float*)d_in[0];   // [N,4]
  const float* scores = (const float*)d_in[1];   // [N]
  float*    ws   = (float*)d_ws;
  uint32_t* mask = (uint32_t*)(ws + 6 * NBOX);
  uint32_t* keep = mask + (size_t)NBOX * WORDS;
  float*    out  = (float*)d_out;

  nms_sort<<<1, 1024, 0, stream>>>(boxes, scores, ws);
  nms_mask<<<dim3(NBOX / 128, WORDS), 128, 0, stream>>>(ws, mask);
  nms_reduce<<<1, 256, 0, stream>>>(mask, keep);
  nms_out<<<NBOX / 256, 256, 0, stream>>>(ws, keep, out);
}